// DTWFramesLayerCUDA_31542239822344
// MI455X (gfx1250) — compile-verified
//
#include <hip/hip_runtime.h>
#include <math.h>

typedef __attribute__((ext_vector_type(2))) float v2f;
typedef __attribute__((ext_vector_type(8))) float v8f;

#define N_X    64
#define N_PATT 32
#define DIM    8
#define W_LEN  128
#define P_LEN  32
#define OUT_W  32
#define WCOEF  1.0f

// One wave32 workgroup per (n, k) pair: fused GEMM (WMMA fp32) -> LDS cost tile
// -> systolic DTW wavefront over 32 lanes -> coalesced output store.
__launch_bounds__(32)
__global__ void dtw_frames_gfx1250_kernel(const float* __restrict__ x,
                                          const float* __restrict__ patts,
                                          float* __restrict__ out) {
  __shared__ float lds_cost[P_LEN][W_LEN];   // 16 KB cost tile
  __shared__ float lds_xx[W_LEN];            // ||x_w||^2
  __shared__ float lds_pp[P_LEN];            // ||patt_p||^2
  __shared__ float lds_out[P_LEN][OUT_W];    // staging for coalesced stores

  const int bid  = blockIdx.x;
  const int n    = bid >> 5;        // / N_PATT
  const int kk   = bid & 31;        // % N_PATT
  const int lane = threadIdx.x;     // 0..31 (wave32)
  const int half = lane >> 4;       // 0/1
  const int l16  = lane & 15;

  const float* __restrict__ xn = x     + (size_t)n  * (DIM * W_LEN);
  const float* __restrict__ pk = patts + (size_t)kk * (DIM * P_LEN);

  // ---------------- norms (cooperative, coalesced) ----------------
  {
    float pp = 0.f;
#pragma unroll
    for (int d = 0; d < DIM; ++d) {
      float v = pk[d * P_LEN + lane];
      pp += v * v;
    }
    lds_pp[lane] = pp;
#pragma unroll
    for (int j = 0; j < 4; ++j) {
      const int w = lane + 32 * j;
      float s = 0.f;
#pragma unroll
      for (int d = 0; d < DIM; ++d) {
        float v = xn[d * W_LEN + w];
        s += v * v;
      }
      lds_xx[w] = s;
    }
  }

  // ---------------- A fragments (patts^T): 16x4 f32 WMMA layout ----------------
  // A[m][kd]: lane L holds M = L%16; VGPR0 -> K = 2*(L/16), VGPR1 -> K = 2*(L/16)+1
  v2f afrag[2][2];
#pragma unroll
  for (int mt = 0; mt < 2; ++mt) {
#pragma unroll
    for (int kc = 0; kc < 2; ++kc) {
      const int d0 = kc * 4 + 2 * half;
      const int m  = mt * 16 + l16;
      afrag[mt][kc].x = pk[(d0 + 0) * P_LEN + m];
      afrag[mt][kc].y = pk[(d0 + 1) * P_LEN + m];
    }
  }

  __syncthreads();

  // ---------------- GEMM: acc[p][w] = sum_d patt[d][p] * x[d][w] ----------------
#pragma unroll
  for (int nt = 0; nt < 8; ++nt) {
    v2f bfrag[2];
#pragma unroll
    for (int kc = 0; kc < 2; ++kc) {
      const int d0 = kc * 4 + 2 * half;
      const int w  = nt * 16 + l16;
      bfrag[kc].x = xn[(d0 + 0) * W_LEN + w];
      bfrag[kc].y = xn[(d0 + 1) * W_LEN + w];
    }
    const float xxw = lds_xx[nt * 16 + l16];
#pragma unroll
    for (int mt = 0; mt < 2; ++mt) {
      v8f acc = {};
#pragma unroll
      for (int kc = 0; kc < 2; ++kc) {
        // 8 args: (neg_a, A, neg_b, B, c_mod, C, reuse_a, reuse_b)
        acc = __builtin_amdgcn_wmma_f32_16x16x4_f32(
            false, afrag[mt][kc], false, bfrag[kc],
            (short)0, acc, false, false);
      }
      // Epilogue: cost = max(||x||^2 + ||p||^2 - 2*dot, 0) into LDS.
      // C/D layout: VGPR r, lane L -> M = r + 8*(L/16), N = L%16.
#pragma unroll
      for (int r = 0; r < 8; ++r) {
        const int p = mt * 16 + half * 8 + r;
        const float c = fmaxf(xxw + lds_pp[p] - 2.0f * acc[r], 0.0f);
        lds_cost[p][nt * 16 + l16] = c;
      }
    }
  }

  __syncthreads();

  // ---------------- systolic DTW: lane p owns row p, lagged p steps ----------------
  // v[p][w] needs: left = v[p][w-1] (own reg), up = v[p-1][w] (shfl of neighbor's
  // last value), diag = v[p-1][w-1] (previous step's 'up').
  const int p = lane;
  float v = 0.f, left = 0.f, diag = 0.f;
  for (int t = 0; t < W_LEN + P_LEN - 1; ++t) {
    const float up = __shfl_up(v, 1, 32);
    const int w = t - p;
    if (w >= 0 && w < W_LEN) {
      const float c = lds_cost[p][w];
      float nv;
      if (p == 0)      nv = (w == 0) ? c : c + WCOEF * left;   // row0 scan
      else if (w == 0) nv = c + up;                            // col0 cumsum
      else             nv = c + fminf(WCOEF * fminf(left, diag), up);
      left = nv;
      v = nv;
      if (w >= W_LEN - OUT_W)
        lds_out[p][w - (W_LEN - OUT_W)] = sqrtf(nv) * (1.0f / (float)P_LEN);
    }
    diag = up;
  }

  __syncthreads();

  // ---------------- coalesced output: 128B per row ----------------
  float* __restrict__ outp = out + (size_t)(n * N_PATT + kk) * P_LEN * OUT_W;
#pragma unroll
  for (int row = 0; row < P_LEN; ++row) {
    outp[row * OUT_W + lane] = lds_out[row][lane];
  }
}

extern "C" void kernel_launch(void* const* d_in, const int* in_sizes, int n_in,
                              void* d_out, int out_size, void* d_ws, size_t ws_size,
                              hipStream_t stream) {
  const float* x     = (const float*)d_in[0];   // (64, 8, 128) fp32
  const float* patts = (const float*)d_in[1];   // (32, 8, 32) fp32
  float* out = (float*)d_out;                   // (64, 32, 32, 32) fp32
  (void)in_sizes; (void)n_in; (void)out_size; (void)d_ws; (void)ws_size;

  dim3 grid(N_X * N_PATT);  // 2048 wave32 workgroups
  dim3 block(32);
  dtw_frames_gfx1250_kernel<<<grid, block, 0, stream>>>(x, patts, out);
}